// Attention_77446850282057
// MI455X (gfx1250) — compile-verified
//
#include <hip/hip_runtime.h>
#include <math.h>

// Problem constants (from reference)
constexpr int NB = 8;       // batch
constexpr int NH = 4;       // heads
constexpr int SL = 256;     // sequence length
constexpr int DKC = 64;     // head dim
constexpr int GNC = 2;      // global heads
constexpr int LNC = 2;      // local heads
constexpr int NDIST = 2 * SL - 1;   // 511
constexpr float MAXV = 10000.0f;

constexpr int QT = 32;       // query rows per block
constexpr int SSTRIDE = 260; // LDS score row stride (floats): no 64-bank conflicts, 8B align kept

typedef __attribute__((ext_vector_type(2))) float v2f;
typedef __attribute__((ext_vector_type(8))) float v8f;

// ---------------------------------------------------------------------------
// Precompute: reldot[n][dist] = dot(rel_pos[dist, n*64:], mlp_w[n])
//             vdot[b][n][k]   = dot(value[b, GN+n, k, :], mlp_w[n])
// ---------------------------------------------------------------------------
__global__ void precompute_kernel(const float* __restrict__ rel_pos,
                                  const float* __restrict__ value,
                                  const float* __restrict__ mlp_w,
                                  float* __restrict__ ws) {
  int tid = blockIdx.x * blockDim.x + threadIdx.x;
  if (tid < LNC * NDIST) {
    int n = tid / NDIST, dist = tid % NDIST;
    const float* r = rel_pos + (size_t)dist * (LNC * DKC) + n * DKC;
    const float* w = mlp_w + n * DKC;
    float acc = 0.f;
#pragma unroll
    for (int d = 0; d < DKC; ++d) acc += r[d] * w[d];
    ws[n * NDIST + dist] = acc;
  } else if (tid < LNC * NDIST + NB * LNC * SL) {
    int idx = tid - LNC * NDIST;
    int b = idx / (LNC * SL);
    int rem = idx % (LNC * SL);
    int n = rem / SL;
    int kk = rem % SL;
    const float* vp = value + (((size_t)(b * NH + GNC + n) * SL + kk) * DKC);
    const float* w = mlp_w + n * DKC;
    float acc = 0.f;
#pragma unroll
    for (int d = 0; d < DKC; ++d) acc += vp[d] * w[d];
    ws[LNC * NDIST + idx] = acc;
  }
}

// ---------------------------------------------------------------------------
// Main fused attention kernel: one block per (b, h, 32-row query tile)
// ---------------------------------------------------------------------------
__global__ __launch_bounds__(256)
void attn_kernel(const float* __restrict__ Q, const float* __restrict__ K,
                 const float* __restrict__ V, const unsigned char* __restrict__ mask,
                 const float* __restrict__ apq, const float* __restrict__ apk,
                 const float* __restrict__ mlp_b,
                 const float* __restrict__ reldot, const float* __restrict__ vdot,
                 float* __restrict__ out, float* __restrict__ pg, float* __restrict__ pl) {
  __shared__ float sS[QT * SSTRIDE];   // 32 x 256 score/prob tile (padded)
  __shared__ float sRed[256];
  __shared__ float sStat[QT];

  const int block = blockIdx.x;
  const int tile = block % (SL / QT);      // 8 query tiles
  const int bh = block / (SL / QT);
  const int b = bh / NH, h = bh % NH;
  const bool isLocal = (h >= GNC);
  const float lmul = isLocal ? 1.0f : 0.0f;   // branch-free local-head flag
  const int n = isLocal ? (h - GNC) : 0;      // n=0 keeps all aux loads in-bounds

  const int tid = threadIdx.x;
  const int lane = tid & 31;          // wave32
  const int w = tid >> 5;             // 8 waves per block
  const int half = lane >> 4;         // 0/1 half-wave
  const int l15 = lane & 15;

  const size_t headOff = (size_t)(b * NH + h) * SL * DKC;
  const float* qptr = Q + headOff;
  const float* kptr = K + headOff;
  const float* vptr = V + headOff;

  const int q0 = tile * QT;
  const int qi = w >> 2;                       // which 16-row sub-tile (0/1)
  const int qrow = q0 + qi * 16 + l15;         // row this lane supplies to A

  // Warm V rows into cache for the PV stage while we do scores.
  __builtin_prefetch(vptr + (size_t)(w * 32 + lane) * DKC, 0, 3);

  // ---- preload Q fragments: 16 K-steps of 4 over DK=64 (branch-free) -----
  v2f qf[16];
#pragma unroll
  for (int s = 0; s < 16; ++s) {
    int d = 4 * s + 2 * half;
    v2f t = *(const v2f*)(qptr + (size_t)qrow * DKC + d);
    v2f a = *(const v2f*)(apq + (size_t)qrow * (LNC * DKC) + n * DKC + d);
    qf[s] = t + lmul * a;
  }

  const float scale = 0.125f;  // 1/sqrt(64)
  const float* vd = vdot + ((size_t)b * LNC + n) * SL;
  const float* rd = reldot + (size_t)n * NDIST;
  const float bias = mlp_b[n];
  const unsigned char* mrow = mask + (size_t)b * SL * SL;

  // ---- scores: S = Q K^T via f32 WMMA; each wave covers 4 key tiles ------
#pragma unroll
  for (int i = 0; i < 4; ++i) {
    const int kt = (w & 3) * 4 + i;
    const int krow = kt * 16 + l15;

    // Batch-load all 16 K fragments first (one wait covers the batch),
    // then issue 16 back-to-back WMMAs.
    v2f kf[16];
#pragma unroll
    for (int s = 0; s < 16; ++s) {
      int d = 4 * s + 2 * half;
      v2f t = *(const v2f*)(kptr + (size_t)krow * DKC + d);
      v2f a = *(const v2f*)(apk + (size_t)krow * (LNC * DKC) + n * DKC + d);
      kf[s] = t + lmul * a;
    }
    v8f acc = {};
#pragma unroll
    for (int s = 0; s < 16; ++s) {
      acc = __builtin_amdgcn_wmma_f32_16x16x4_f32(false, qf[s], false, kf[s],
                                                  (short)0, acc, false, false);
    }
    // post-process tile and spill to LDS (mask select is VALU cndmask)
#pragma unroll
    for (int r = 0; r < 8; ++r) {
      int ql = qi * 16 + r + 8 * half;   // local row in [0,32)
      int qg = q0 + ql;
      int kg = kt * 16 + l15;
      float sc = acc[r] * scale;
      sc += lmul * (vd[qg] + vd[kg] + rd[kg - qg + (SL - 1)] + bias);
      if (mrow[(size_t)qg * SL + kg]) sc = -MAXV;
      sS[ql * SSTRIDE + kg] = sc;
    }
  }
  __syncthreads();

  // ---- softmax across each of the 32 rows (8 threads / row) --------------
  {
    int r = tid >> 3;
    int seg = tid & 7;
    float* row = sS + r * SSTRIDE + seg * 32;
    float m = -3.0e38f;
#pragma unroll
    for (int c = 0; c < 32; ++c) m = fmaxf(m, row[c]);
    sRed[tid] = m;
    __syncthreads();
    if (tid < QT) {
      float m2 = sRed[tid * 8];
#pragma unroll
      for (int j = 1; j < 8; ++j) m2 = fmaxf(m2, sRed[tid * 8 + j]);
      sStat[tid] = m2;
    }
    __syncthreads();
    float rmax = sStat[r];
    float sum = 0.f;
#pragma unroll
    for (int c = 0; c < 32; ++c) {
      float e = __expf(row[c] - rmax);
      row[c] = e;
      sum += e;
    }
    sRed[tid] = sum;
    __syncthreads();
    if (tid < QT) {
      float s2 = 0.f;
#pragma unroll
      for (int j = 0; j < 8; ++j) s2 += sRed[tid * 8 + j];
      sStat[tid] = 1.f / s2;
    }
    __syncthreads();
    float rinv = sStat[r];
    int qg = q0 + r;
    float* pbase = isLocal
        ? (pl + (((size_t)b * LNC + n) * SL + qg) * SL + seg * 32)
        : (pg + (((size_t)b * GNC + h) * SL + qg) * SL + seg * 32);
#pragma unroll
    for (int c = 0; c < 32; ++c) {
      float p = row[c] * rinv;
      row[c] = p;       // keep normalized probs in LDS for PV
      pbase[c] = p;     // and emit p_g / p_l
    }
  }
  __syncthreads();

  // ---- O = P * V : each wave owns one 16x16 tile of the 32x64 output -----
  // 64 K-steps of 4, processed in chunks of 8: batch loads, then 8 WMMAs.
  {
    const int qi2 = w >> 2;   // 0/1
    const int di = w & 3;     // 0..3
    const int d = di * 16 + l15;
    const float* arow = sS + (qi2 * 16 + l15) * SSTRIDE;
    v8f acc = {};
    for (int c8 = 0; c8 < 8; ++c8) {
      v2f af[8], bf[8];
#pragma unroll
      for (int j = 0; j < 8; ++j) {
        int s = c8 * 8 + j;
        int kk = 4 * s + 2 * half;
        af[j] = *(const v2f*)(arow + kk);
        v2f t;
        t.x = vptr[(size_t)kk * DKC + d];
        t.y = vptr[(size_t)(kk + 1) * DKC + d];
        bf[j] = t;
      }
#pragma unroll
      for (int j = 0; j < 8; ++j) {
        acc = __builtin_amdgcn_wmma_f32_16x16x4_f32(false, af[j], false, bf[j],
                                                    (short)0, acc, false, false);
      }
    }
#pragma unroll
    for (int r = 0; r < 8; ++r) {
      int qg = q0 + qi2 * 16 + r + 8 * half;
      out[headOff + (size_t)qg * DKC + d] = acc[r];
    }
  }
}

// ---------------------------------------------------------------------------
extern "C" void kernel_launch(void* const* d_in, const int* in_sizes, int n_in,
                              void* d_out, int out_size, void* d_ws, size_t ws_size,
                              hipStream_t stream) {
  const float* q = (const float*)d_in[0];
  const float* k = (const float*)d_in[1];
  const float* v = (const float*)d_in[2];
  const unsigned char* mask = (const unsigned char*)d_in[3];  // jax bool = 1 byte
  const float* apq = (const float*)d_in[4];
  const float* apk = (const float*)d_in[5];
  const float* rel = (const float*)d_in[6];
  const float* mw = (const float*)d_in[7];
  const float* mb = (const float*)d_in[8];

  float* out = (float*)d_out;
  float* pg = out + (size_t)NB * NH * SL * DKC;
  float* pl = pg + (size_t)NB * GNC * SL * SL;
  float* ws = (float*)d_ws;

  int pre_n = LNC * NDIST + NB * LNC * SL;
  precompute_kernel<<<(pre_n + 255) / 256, 256, 0, stream>>>(rel, v, mw, ws);

  int grid = NB * NH * (SL / QT);  // 256 blocks
  attn_kernel<<<grid, 256, 0, stream>>>(q, k, v, mask, apq, apk, mb,
                                        ws, ws + LNC * NDIST, out, pg, pl);
}